// QuestionRNN_34935263985732
// MI455X (gfx1250) — compile-verified
//
#include <hip/hip_runtime.h>
#include <hip/hip_bf16.h>

#define VOCAB 128000
#define HID   256
#define SEQ   512

typedef float v2f __attribute__((ext_vector_type(2)));
typedef float v8f __attribute__((ext_vector_type(8)));

// ---------------------------------------------------------------------------
// Phase 1: inp[s][n] = sum_k emb[idx[s]][k] * W[n][HID + k]  + b[n]
// One wave per 16x16 output tile, V_WMMA_F32_16X16X4_F32, K-loop step 4.
// (Verified round 1: emits v_wmma_f32_16x16x4_f32 with clause-grouped b64
//  fragment loads. Off the critical path: ~67 MFLOP, all inputs < 1.5 MB.)
// ---------------------------------------------------------------------------
__global__ __launch_bounds__(256)
void rnn_inp_gemm(const int* __restrict__ idx,
                  const float* __restrict__ emb,
                  const float* __restrict__ W,
                  const float* __restrict__ bias,
                  float* __restrict__ inp) {
    const int wave = threadIdx.x >> 5;          // 8 waves per block
    const int lane = threadIdx.x & 31;
    const int tile = blockIdx.x * 8 + wave;     // (SEQ/16)*(HID/16) = 512 tiles
    const int tiles_n = HID / 16;               // 16
    const int s0 = (tile / tiles_n) * 16;       // seq-tile origin
    const int n0 = (tile % tiles_n) * 16;       // hidden-tile origin

    const int half = lane >> 4;                 // 0: lanes 0-15, 1: lanes 16-31
    const int mn   = lane & 15;                 // M for A frag, N for B/C/D frags
    const int kb   = half * 2;                  // A/B K base: {0,1} or {2,3}

    // A: 16x4 fp32 tile of gathered embedding rows.
    const long arow = (long)idx[s0 + mn];
    const float* __restrict__ aptr = emb + arow * (long)HID;
    // B: B[k][n] = Wx[n][k] = W[(n0+n)*2*HID + HID + k]
    const float* __restrict__ bptr = W + (long)(n0 + mn) * (2 * HID) + HID;

    v8f c = {};
    #pragma unroll 4
    for (int k0 = 0; k0 < HID; k0 += 4) {
        v2f a, bm;
        a.x  = aptr[k0 + kb + 0];
        a.y  = aptr[k0 + kb + 1];
        bm.x = bptr[k0 + kb + 0];
        bm.y = bptr[k0 + kb + 1];
        c = __builtin_amdgcn_wmma_f32_16x16x4_f32(
                /*neg_a=*/false, a, /*neg_b=*/false, bm,
                /*c_mod=*/(short)0, c, /*reuse_a=*/false, /*reuse_b=*/false);
    }

    const float bv = bias[n0 + mn];
    #pragma unroll
    for (int r = 0; r < 8; ++r) {
        const int m = half * 8 + r;
        inp[(long)(s0 + m) * HID + n0 + mn] = c[r] + bv;
    }
}

// ---------------------------------------------------------------------------
// Phase 2: sequential scan  h = tanh(h @ Wh.T + inp[t]),  512 steps.
//
// Register-resident weights: 512 threads, thread (t = tid&255, half = tid>>8)
// owns Wh[t][half*128 .. half*128+127] in 64 packed float2 VGPRs. Per step:
//   - broadcast-read h (same-address b128 LDS loads, ~free),
//   - 64 v_pk_fma_f32 into 4 rotating accumulators (breaks the FMA chain),
//   - 512-float partial-sum exchange through LDS,
//   - threads < 256 combine + tanh + publish new h.
// LDS traffic per step drops from 256 KB (round 1) to ~3 KB.
// ---------------------------------------------------------------------------
__global__ __launch_bounds__(512)
void rnn_scan(const float* __restrict__ W,
              const float* __restrict__ inp,
              float* __restrict__ out) {
    __shared__ float hbuf[HID];        // current hidden state
    __shared__ float psum[2 * HID];    // per-thread partial dot products

    const int tid  = threadIdx.x;      // 0..511
    const int t    = tid & (HID - 1);  // output element this thread serves
    const int kofs = (tid >> 8) * (HID / 2);   // k-range origin: 0 or 128

    // Load this thread's 128 weights Wh[t][kofs..kofs+127] into VGPRs once.
    v2f wreg[64];
    const float* __restrict__ wrow = W + (long)t * (2 * HID) + kofs;
    #pragma unroll
    for (int i = 0; i < 32; ++i) {
        const float4 w4 = *(const float4*)(wrow + 4 * i);
        wreg[2 * i + 0] = (v2f){w4.x, w4.y};
        wreg[2 * i + 1] = (v2f){w4.z, w4.w};
    }

    if (tid < HID) hbuf[tid] = 0.0f;
    __syncthreads();

    const float* __restrict__ hp = hbuf + kofs;

    for (int step = 0; step < SEQ; ++step) {
        // Early-issue the inp row load (L2-hot); consumed after the barrier.
        float inp_v = 0.0f;
        if (tid < HID) inp_v = inp[(long)step * HID + tid];

        v2f acc[4] = {};
        #pragma unroll
        for (int i = 0; i < 32; ++i) {
            const float4 h4 = *(const float4*)(hp + 4 * i);   // broadcast b128
            const v2f hlo = (v2f){h4.x, h4.y};
            const v2f hhi = (v2f){h4.z, h4.w};
            acc[(2 * i + 0) & 3] =
                __builtin_elementwise_fma(hlo, wreg[2 * i + 0], acc[(2 * i + 0) & 3]);
            acc[(2 * i + 1) & 3] =
                __builtin_elementwise_fma(hhi, wreg[2 * i + 1], acc[(2 * i + 1) & 3]);
        }
        const v2f s = (acc[0] + acc[1]) + (acc[2] + acc[3]);
        psum[tid] = s.x + s.y;
        __syncthreads();                       // partials published, h reads done

        if (tid < HID) {
            const float a = psum[tid] + psum[tid + HID] + inp_v;
            hbuf[tid] = tanhf(a);
        }
        __syncthreads();                       // new h visible to all
    }

    if (tid < HID) out[tid] = hbuf[tid];
}

// ---------------------------------------------------------------------------
extern "C" void kernel_launch(void* const* d_in, const int* in_sizes, int n_in,
                              void* d_out, int out_size, void* d_ws, size_t ws_size,
                              hipStream_t stream) {
    const int*   idx = (const int*)  d_in[0];   // (SEQ,)
    const float* emb = (const float*)d_in[1];   // (VOCAB, HID)
    const float* W   = (const float*)d_in[2];   // (HID, 2*HID)
    const float* b   = (const float*)d_in[3];   // (HID,)
    float* out = (float*)d_out;                 // (HID,)
    float* inp = (float*)d_ws;                  // SEQ*HID floats = 512 KB scratch

    rnn_inp_gemm<<<64, 256, 0, stream>>>(idx, emb, W, b, inp);
    rnn_scan<<<1, 512, 0, stream>>>(W, inp, out);
}